// MultiPersonHeatmapOffsetAggregationLoss_62551903699685
// MI455X (gfx1250) — compile-verified
//
#include <hip/hip_runtime.h>

// ---------------------------------------------------------------------------
// MultiPersonHeatmapOffsetAggregationLoss for MI455X (gfx1250, wave32).
//
// loss = 4 * ( mean(softplus(pred)) - sum_q (P-q)*s_q / (P*N) ) + 0.02
//   s_q = sum over valid keypoints k of pred[q,k, disk(radius=3) around kp]
//
// Memory-bound: one 66.8 MB pass over pred_heatmaps (~2.9 us at 23.3 TB/s).
// Trans work minimized via log fusion:
//   sum_i ln(1+e_i) = ln(prod_i (1+e_i)),  (1+e_i) in [1,2]
// -> 1.25 trans ops/element (4x v_exp_f32 + 1x v_log_f32 per float4),
// keeping the trans pipe under the HBM floor. Running sums accumulate on the
// matrix pipe with V_WMMA_F32_16X16X4_F32 (A = packed quad-softplus sums,
// B = ones). Deterministic two-pass reduction (no float atomics).
// ---------------------------------------------------------------------------

typedef __attribute__((ext_vector_type(2))) float v2f;
typedef __attribute__((ext_vector_type(8))) float v8f;

#define NBLK 2048
#define TPB  256
#define P_POSES 15
#define K_KPTS  17
#define HW_DIM  256

// Sum of softplus over a float4:
//   sum = sum(max(x,0)) + ln2 * log2( prod(1 + 2^(-|x|*log2e)) )
// Product is in [1,16]; raw v_exp_f32/v_log_f32, no denorm fixups possible.
__device__ __forceinline__ float softplus_quad(float4 v) {
    const float kNegLog2e = -1.4426950408889634f;
    float t0 = __builtin_amdgcn_exp2f(__builtin_fabsf(v.x) * kNegLog2e);
    float t1 = __builtin_amdgcn_exp2f(__builtin_fabsf(v.y) * kNegLog2e);
    float t2 = __builtin_amdgcn_exp2f(__builtin_fabsf(v.z) * kNegLog2e);
    float t3 = __builtin_amdgcn_exp2f(__builtin_fabsf(v.w) * kNegLog2e);
    float p  = ((1.0f + t0) * (1.0f + t1)) * ((1.0f + t2) * (1.0f + t3));
    float m  = (fmaxf(v.x, 0.0f) + fmaxf(v.y, 0.0f)) +
               (fmaxf(v.z, 0.0f) + fmaxf(v.w, 0.0f));
    return __builtin_fmaf(0.6931471805599453f, __builtin_amdgcn_logf(p), m);
}

__device__ __forceinline__ v8f wmma_acc(v2f a, v2f b, v8f c) {
    return __builtin_amdgcn_wmma_f32_16x16x4_f32(
        /*neg_a=*/false, a, /*neg_b=*/false, b,
        /*c_mod=*/(short)0, c, /*reuse_a=*/false, /*reuse_b=*/false);
}

// Kernel A: per-block partial sums of softplus(pred), WMMA-accumulated.
__global__ void softplus_partial_kernel(const float* __restrict__ pred,
                                        float* __restrict__ partial, int n4) {
    const float4* p4 = (const float4*)pred;
    const int stride = gridDim.x * blockDim.x;
    const int base   = blockIdx.x * blockDim.x + threadIdx.x;
    const int pairs  = n4 / (2 * stride);   // iterations with both sweeps full

    v8f c = {0.f, 0.f, 0.f, 0.f, 0.f, 0.f, 0.f, 0.f};
    const v2f bones = {1.0f, 1.0f};         // B = all ones (4x16)

    // Branchless hot loop: 2x b128 loads, 8 elements/lane, 1 WMMA, full EXEC.
    for (int it = 0; it < pairs; ++it) {
        float4 v0 = p4[base + (2 * it + 0) * stride];
        float4 v1 = p4[base + (2 * it + 1) * stride];
        v2f a;
        a.x = softplus_quad(v0);
        a.y = softplus_quad(v1);
        c = wmma_acc(a, bones, c);          // D = A x ones + C on the XDL pipe
    }

    // Up to two predicated tail sweeps (EXEC restored before the WMMA).
    {
        int i0 = base + (2 * pairs + 0) * stride;
        int i1 = base + (2 * pairs + 1) * stride;
        v2f a = {0.0f, 0.0f};
        if (i0 < n4) a.x = softplus_quad(p4[i0]);
        if (i1 < n4) a.y = softplus_quad(p4[i1]);
        c = wmma_acc(a, bones, c);
    }

    // lanes 0-15 hold sum of D rows 0..7, lanes 16-31 hold rows 8..15
    float s = c[0] + c[1] + c[2] + c[3] + c[4] + c[5] + c[6] + c[7];
    s += __shfl_xor(s, 16, 32);             // full wave total, all lanes

    __shared__ float wsum[TPB / 32];
    if ((threadIdx.x & 31) == 0) wsum[threadIdx.x >> 5] = s;
    __syncthreads();
    if (threadIdx.x == 0) {
        float tot = 0.0f;
        for (int w = 0; w < TPB / 32; ++w) tot += wsum[w];   // fixed order
        partial[blockIdx.x] = tot;
    }
}

// Kernel B: per-keypoint weighted disk sums (255 keypoints, <=29 px each).
__global__ void keypoint_kernel(const float* __restrict__ pred,
                                const float* __restrict__ tk,
                                float* __restrict__ out) {
    int t = threadIdx.x;                    // 0..255
    float r = 0.0f;
    if (t < P_POSES * K_KPTS) {
        int p = t / K_KPTS;
        float x = tk[2 * t + 0];
        float y = tk[2 * t + 1];
        bool valid = ((x != 0.0f) && (x != -1.0f)) || ((y != 0.0f) && (y != -1.0f));
        if (valid) {
            int xi = (int)x;                // trunc, matches astype(int32)
            int yi = (int)y;
            const float* img = pred + (size_t)t * HW_DIM * HW_DIM;
            float s = 0.0f;
            #pragma unroll
            for (int dy = -3; dy <= 3; ++dy) {
                int yy = yi + dy;
                if (yy < 0 || yy >= HW_DIM) continue;
                #pragma unroll
                for (int dx = -3; dx <= 3; ++dx) {
                    int xx = xi + dx;
                    if (xx < 0 || xx >= HW_DIM) continue;
                    if (dy * dy + dx * dx <= 9)
                        s += img[yy * HW_DIM + xx];
                }
            }
            r = (float)(P_POSES - p) * s;   // pose q reused (P-q) times
        }
    }
    out[t] = r;
}

// Kernel C: deterministic fixed-order reduce of both partial arrays + combine.
__global__ void finalize_kernel(const float* __restrict__ W,
                                float* __restrict__ out) {
    __shared__ float s1[TPB];
    __shared__ float s2[TPB];
    int t = threadIdx.x;
    float a = 0.0f;
    for (int i = t; i < NBLK; i += TPB) a += W[i];   // fixed order per slot
    s1[t] = a;
    s2[t] = W[NBLK + t];
    __syncthreads();
    for (int off = TPB / 2; off > 0; off >>= 1) {
        if (t < off) { s1[t] += s1[t + off]; s2[t] += s2[t + off]; }
        __syncthreads();
    }
    if (t == 0) {
        const double Nd = (double)P_POSES * K_KPTS * HW_DIM * HW_DIM;
        double hl = (double)s1[0] / Nd - (double)s2[0] / ((double)P_POSES * Nd);
        out[0] = (float)(4.0 * hl + 0.02);
    }
}

extern "C" void kernel_launch(void* const* d_in, const int* in_sizes, int n_in,
                              void* d_out, int out_size, void* d_ws, size_t ws_size,
                              hipStream_t stream) {
    const float* pred = (const float*)d_in[0];       // [15,17,256,256]
    const float* tk   = (const float*)d_in[2];       // [15,17,2]
    float* W = (float*)d_ws;                         // [NBLK] + [256] partials

    int n  = in_sizes[0];                            // 16,711,680 (div by 4)
    int n4 = n / 4;

    softplus_partial_kernel<<<NBLK, TPB, 0, stream>>>(pred, W, n4);
    keypoint_kernel<<<1, TPB, 0, stream>>>(pred, tk, W + NBLK);
    finalize_kernel<<<1, TPB, 0, stream>>>(W, (float*)d_out);
}